// MolDeeperGCN_30434138259681
// MI455X (gfx1250) — compile-verified
//
#include <hip/hip_runtime.h>
#include <math.h>

#define NN 100000
#define EE 1600000
#define HH 64
#define LL 4
#define GG 512
#define TT 12
#define ATOMF 9
#define ATOMV 128
#define BONDF 3
#define BONDV 16
#define BN_EPS 1e-5f
#define GEN_EPS 1e-7f

typedef __attribute__((ext_vector_type(2))) float v2f;
typedef __attribute__((ext_vector_type(8))) float v8f;

// ---- monotonic float<->int mapping for atomicMax on floats ----
__device__ __forceinline__ int f2ord(float f) {
    int i = __float_as_int(f);
    return (i >= 0) ? i : (i ^ 0x7FFFFFFF);
}
__device__ __forceinline__ float ord2f(int i) {
    return __int_as_float((i >= 0) ? i : (i ^ 0x7FFFFFFF));
}

// ---------------- atom encoder: h[n,c] = sum_f atom_emb[f, x[n,f], c] ----------------
__global__ void k_atom_enc(const int* __restrict__ x, const float* __restrict__ aemb,
                           float* __restrict__ h) {
    int idx = blockIdx.x * blockDim.x + threadIdx.x;
    if (idx >= NN * HH) return;
    int n = idx >> 6, c = idx & 63;
    float s = 0.f;
#pragma unroll
    for (int f = 0; f < ATOMF; ++f) {
        int v = x[n * ATOMF + f];
        s += aemb[(f * ATOMV + v) * HH + c];
    }
    h[idx] = s;
}

// ---------------- BN stats: sums[0:64]=sum, sums[64:128]=sumsq ----------------
__global__ void k_bn_stats(const float* __restrict__ h, float* __restrict__ sums) {
    __shared__ float s1[256], s2[256];
    int c = threadIdx.x & 63;
    int rowg = threadIdx.x >> 6;  // 0..3
    float a = 0.f, b = 0.f;
    for (int n = blockIdx.x * 4 + rowg; n < NN; n += gridDim.x * 4) {
        float v = h[n * HH + c];
        a += v;
        b += v * v;
    }
    s1[threadIdx.x] = a;
    s2[threadIdx.x] = b;
    __syncthreads();
    if (threadIdx.x < 64) {
        float ta = s1[c] + s1[64 + c] + s1[128 + c] + s1[192 + c];
        float tb = s2[c] + s2[64 + c] + s2[128 + c] + s2[192 + c];
        atomicAdd(&sums[c], ta);
        atomicAdd(&sums[64 + c], tb);
    }
}

// ---------------- BN apply + ReLU ----------------
__global__ void k_bn_relu(const float* __restrict__ h, const float* __restrict__ sums,
                          const float* __restrict__ gamma, const float* __restrict__ beta,
                          float* __restrict__ hn) {
    int idx = blockIdx.x * blockDim.x + threadIdx.x;
    if (idx >= NN * HH) return;
    int c = idx & 63;
    float mu = sums[c] * (1.f / NN);
    float var = sums[64 + c] * (1.f / NN) - mu * mu;
    float r = rsqrtf(var + BN_EPS);
    float v = gamma[c] * (h[idx] - mu) * r + beta[c];
    hn[idx] = fmaxf(v, 0.f);
}

// ---------------- init segment buffers ----------------
__global__ void k_init_seg(int* __restrict__ mk, float* __restrict__ den,
                           float* __restrict__ num) {
    int idx = blockIdx.x * blockDim.x + threadIdx.x;
    if (idx >= NN * HH) return;
    mk[idx] = (int)0x80000000;  // INT_MIN (< f2ord of anything)
    den[idx] = 0.f;
    num[idx] = 0.f;
}

__global__ void k_zero(float* __restrict__ p, int n) {
    int idx = blockIdx.x * blockDim.x + threadIdx.x;
    if (idx < n) p[idx] = 0.f;
}

// ---------------- edge pass A: segment max of z = t*(relu(h[src]+e)+eps) ----------------
// 16 threads/edge, float4 channels; bond table for this layer staged in LDS.
__global__ void __launch_bounds__(256) k_edge_max(
    const float* __restrict__ hn, const int* __restrict__ ei, const int* __restrict__ eattr,
    const float* __restrict__ bemb, const float* __restrict__ tAll, int layer,
    int* __restrict__ mk) {
    __shared__ float blds[BONDF * BONDV * HH];
    const float* bl = bemb + (size_t)layer * BONDF * BONDV * HH;
    for (int i = threadIdx.x; i < BONDF * BONDV * HH; i += blockDim.x) blds[i] = bl[i];
    __syncthreads();
    float t = tAll[layer];
    int e = blockIdx.x * 16 + (threadIdx.x >> 4);
    int c = (threadIdx.x & 15) * 4;
    if (e >= EE) return;
    int src = ei[e], dst = ei[EE + e];
    int a0 = eattr[e * 3 + 0], a1 = eattr[e * 3 + 1], a2 = eattr[e * 3 + 2];
    float4 hs = *(const float4*)(hn + (size_t)src * HH + c);
    float4 e0 = *(const float4*)(blds + (0 * BONDV + a0) * HH + c);
    float4 e1 = *(const float4*)(blds + (1 * BONDV + a1) * HH + c);
    float4 e2 = *(const float4*)(blds + (2 * BONDV + a2) * HH + c);
    float s[4] = {hs.x + e0.x + e1.x + e2.x, hs.y + e0.y + e1.y + e2.y,
                  hs.z + e0.z + e1.z + e2.z, hs.w + e0.w + e1.w + e2.w};
    int* mrow = mk + (size_t)dst * HH + c;
#pragma unroll
    for (int j = 0; j < 4; ++j) {
        float msg = fmaxf(s[j], 0.f) + GEN_EPS;
        atomicMax(&mrow[j], f2ord(t * msg));
    }
}

// ---------------- edge pass B: den += exp(z-m), num += exp(z-m)*msg ----------------
__global__ void __launch_bounds__(256) k_edge_sum(
    const float* __restrict__ hn, const int* __restrict__ ei, const int* __restrict__ eattr,
    const float* __restrict__ bemb, const float* __restrict__ tAll, int layer,
    const int* __restrict__ mk, float* __restrict__ den, float* __restrict__ num) {
    __shared__ float blds[BONDF * BONDV * HH];
    const float* bl = bemb + (size_t)layer * BONDF * BONDV * HH;
    for (int i = threadIdx.x; i < BONDF * BONDV * HH; i += blockDim.x) blds[i] = bl[i];
    __syncthreads();
    float t = tAll[layer];
    int e = blockIdx.x * 16 + (threadIdx.x >> 4);
    int c = (threadIdx.x & 15) * 4;
    if (e >= EE) return;
    int src = ei[e], dst = ei[EE + e];
    int a0 = eattr[e * 3 + 0], a1 = eattr[e * 3 + 1], a2 = eattr[e * 3 + 2];
    float4 hs = *(const float4*)(hn + (size_t)src * HH + c);
    float4 e0 = *(const float4*)(blds + (0 * BONDV + a0) * HH + c);
    float4 e1 = *(const float4*)(blds + (1 * BONDV + a1) * HH + c);
    float4 e2 = *(const float4*)(blds + (2 * BONDV + a2) * HH + c);
    float s[4] = {hs.x + e0.x + e1.x + e2.x, hs.y + e0.y + e1.y + e2.y,
                  hs.z + e0.z + e1.z + e2.z, hs.w + e0.w + e1.w + e2.w};
    size_t base = (size_t)dst * HH + c;
#pragma unroll
    for (int j = 0; j < 4; ++j) {
        float msg = fmaxf(s[j], 0.f) + GEN_EPS;
        float z = t * msg;
        float mv = ord2f(mk[base + j]);
        float ex = __expf(z - mv);
        atomicAdd(&den[base + j], ex);
        atomicAdd(&num[base + j], ex * msg);
    }
}

// ---------------- combine: y = num/max(den,1e-16) + hn ----------------
__global__ void k_combine(const float* __restrict__ num, const float* __restrict__ den,
                          const float* __restrict__ hn, float* __restrict__ y) {
    int idx = blockIdx.x * blockDim.x + threadIdx.x;
    if (idx >= NN * HH) return;
    float agg = num[idx] / fmaxf(den[idx], 1e-16f);
    y[idx] = agg + hn[idx];
}

// ---------------- WMMA GEMM: Hout[n,:] (+)= X[n,:] @ W + b ----------------
// 4 waves/block, each wave does 16 rows x 64 cols via V_WMMA_F32_16X16X4_F32.
// W is staged in LDS *pre-swizzled by K-pair* so each lane's B operand
// (W[kk][col], W[kk+1][col]) is one contiguous 8-byte v2f -> single ds_load_b64
// into an even-aligned VGPR pair (no v_mov shuffles in the inner loop).
__global__ void __launch_bounds__(128) k_gemm_wmma(
    const float* __restrict__ X, const float* __restrict__ W, const float* __restrict__ bias,
    float* __restrict__ Hout, int residual) {
    __shared__ float wswz[HH * HH];  // [kpair(32)][col(64)][r(2)]
    __shared__ float blv[HH];
    for (int i = threadIdx.x; i < HH * HH; i += 128) {
        int k = i >> 6;      // source row
        int col = i & 63;    // source col
        wswz[(((k >> 1) * HH) + col) * 2 + (k & 1)] = W[i];
    }
    if (threadIdx.x < HH) blv[threadIdx.x] = bias[threadIdx.x];
    __syncthreads();
    int wave = threadIdx.x >> 5;
    int lane = threadIdx.x & 31;
    int half = lane >> 4;   // K sub-pair selector
    int lrow = lane & 15;   // M for A, N for B/C/D
    int rowBase = blockIdx.x * 64 + wave * 16;
    if (rowBase >= NN) return;  // N % 16 == 0, uniform per wave => EXEC stays all-1s

    v8f acc[4] = {};
    const float* xrow = X + (size_t)(rowBase + lrow) * HH;
#pragma unroll
    for (int k0 = 0; k0 < HH; k0 += 4) {
        int kk = k0 + 2 * half;
        int kp = kk >> 1;  // K-pair index
        v2f a;
        a.x = xrow[kk];
        a.y = xrow[kk + 1];
#pragma unroll
        for (int nt = 0; nt < 4; ++nt) {
            v2f b = *(const v2f*)(wswz + ((kp * HH) + nt * 16 + lrow) * 2);
            acc[nt] = __builtin_amdgcn_wmma_f32_16x16x4_f32(
                false, a, false, b, (short)0, acc[nt], false, false);
        }
    }
#pragma unroll
    for (int nt = 0; nt < 4; ++nt) {
#pragma unroll
        for (int j = 0; j < 8; ++j) {
            int row = rowBase + j + 8 * half;
            int col = nt * 16 + lrow;
            float v = acc[nt][j] + blv[col];
            if (residual) v += Hout[(size_t)row * HH + col];
            Hout[(size_t)row * HH + col] = v;
        }
    }
}

// ---------------- mean pool (atomics) ----------------
__global__ void k_pool(const float* __restrict__ hn, const int* __restrict__ batch,
                       float* __restrict__ pool, float* __restrict__ cnt) {
    int idx = blockIdx.x * blockDim.x + threadIdx.x;
    if (idx >= NN * HH) return;
    int n = idx >> 6, c = idx & 63;
    int g = batch[n];
    atomicAdd(&pool[g * HH + c], hn[idx]);
    if (c == 0) atomicAdd(&cnt[g], 1.f);
}

// ---------------- head: out[g,t] = (pool[g,:]/max(cnt,1)) @ lin_W + lin_b ----------------
__global__ void k_head(const float* __restrict__ pool, const float* __restrict__ cnt,
                       const float* __restrict__ lw, const float* __restrict__ lb,
                       float* __restrict__ out) {
    int idx = blockIdx.x * blockDim.x + threadIdx.x;
    if (idx >= GG * TT) return;
    int g = idx / TT, tt = idx % TT;
    float inv = 1.f / fmaxf(cnt[g], 1.f);
    float s = lb[tt];
#pragma unroll
    for (int c = 0; c < HH; ++c) s += pool[g * HH + c] * inv * lw[c * TT + tt];
    out[idx] = s;
}

extern "C" void kernel_launch(void* const* d_in, const int* in_sizes, int n_in,
                              void* d_out, int out_size, void* d_ws, size_t ws_size,
                              hipStream_t stream) {
    const int* x      = (const int*)d_in[0];
    const int* eattr  = (const int*)d_in[1];
    const int* ei     = (const int*)d_in[2];
    const int* batch  = (const int*)d_in[3];
    const float* aemb = (const float*)d_in[4];
    const float* bemb = (const float*)d_in[5];
    const float* mlpW = (const float*)d_in[6];
    const float* mlpb = (const float*)d_in[7];
    const float* tArr = (const float*)d_in[8];
    const float* bngm = (const float*)d_in[9];
    const float* bnbt = (const float*)d_in[10];
    const float* linW = (const float*)d_in[11];
    const float* linb = (const float*)d_in[12];
    float* out = (float*)d_out;

    // workspace carve-up
    char* w = (char*)d_ws;
    const size_t NH = (size_t)NN * HH;
    float* h   = (float*)w;              w += NH * sizeof(float);
    float* hn  = (float*)w;              w += NH * sizeof(float);
    int*   mk  = (int*)w;                w += NH * sizeof(int);
    float* den = (float*)w;              w += NH * sizeof(float);
    float* num = (float*)w;              w += NH * sizeof(float);
    float* y   = (float*)w;              w += NH * sizeof(float);
    float* bns = (float*)w;              w += 128 * sizeof(float);
    float* pool = (float*)w;             w += (size_t)GG * HH * sizeof(float);
    float* cnt  = (float*)w;             /* GG floats (pool & cnt contiguous) */

    const int NHtot = NN * HH;
    const int gNH = (NHtot + 255) / 256;
    const int gE  = (EE + 15) / 16;
    const int gGemm = (NN + 63) / 64;

    k_atom_enc<<<gNH, 256, 0, stream>>>(x, aemb, h);

    for (int l = 0; l < LL; ++l) {
        const float* hin = h;
        if (l > 0) {
            k_zero<<<1, 128, 0, stream>>>(bns, 128);
            k_bn_stats<<<1024, 256, 0, stream>>>(h, bns);
            k_bn_relu<<<gNH, 256, 0, stream>>>(h, bns, bngm + l * HH, bnbt + l * HH, hn);
            hin = hn;
        }
        k_init_seg<<<gNH, 256, 0, stream>>>(mk, den, num);
        k_edge_max<<<gE, 256, 0, stream>>>(hin, ei, eattr, bemb, tArr, l, mk);
        k_edge_sum<<<gE, 256, 0, stream>>>(hin, ei, eattr, bemb, tArr, l, mk, den, num);
        k_combine<<<gNH, 256, 0, stream>>>(num, den, hin, y);
        k_gemm_wmma<<<gGemm, 128, 0, stream>>>(y, mlpW + (size_t)l * HH * HH, mlpb + l * HH,
                                               h, l > 0 ? 1 : 0);
    }

    // final norm+act (uses layer-0 gamma/beta), pool, head
    k_zero<<<1, 128, 0, stream>>>(bns, 128);
    k_bn_stats<<<1024, 256, 0, stream>>>(h, bns);
    k_bn_relu<<<gNH, 256, 0, stream>>>(h, bns, bngm, bnbt, hn);
    k_zero<<<(GG * HH + GG + 255) / 256, 256, 0, stream>>>(pool, GG * HH + GG);
    k_pool<<<gNH, 256, 0, stream>>>(hn, batch, pool, cnt);
    k_head<<<(GG * TT + 255) / 256, 256, 0, stream>>>(pool, cnt, linW, linb, out);
}